// TAGCN_32899449488058
// MI455X (gfx1250) — compile-verified
//
#include <hip/hip_runtime.h>
#include <utility>

typedef float v2f __attribute__((ext_vector_type(2)));
typedef float v8f __attribute__((ext_vector_type(8)));
typedef unsigned int u32x4 __attribute__((ext_vector_type(4)));
typedef int i32x4 __attribute__((ext_vector_type(4)));
typedef int i32x8 __attribute__((ext_vector_type(8)));

#define LN_EPS 1e-5f

// ---------------------------------------------------------------------------
// Elementwise helpers
// ---------------------------------------------------------------------------
__global__ void zero_kernel(float* __restrict__ p, int n) {
    int i = blockIdx.x * blockDim.x + threadIdx.x;
    if (i < n) p[i] = 0.0f;
}

__global__ void deg_kernel(const int* __restrict__ dst, float* __restrict__ deg, int E) {
    int e = blockIdx.x * blockDim.x + threadIdx.x;
    if (e < E) atomicAdd(&deg[dst[e]], 1.0f);
}

__global__ void norm_finalize_kernel(float* __restrict__ norm, int N) {
    int i = blockIdx.x * blockDim.x + threadIdx.x;
    if (i < N) norm[i] = rsqrtf(fmaxf(norm[i], 1.0f));
}

// acc[i, f] *= norm[i]   (row-major [N, 96])
__global__ void row_scale_kernel(float* __restrict__ acc, const float* __restrict__ norm, int total) {
    int i = blockIdx.x * blockDim.x + threadIdx.x;
    if (i < total) acc[i] *= norm[i / 96];
}

// ---------------------------------------------------------------------------
// Edge scatter: acc[dst] += r[src] * norm[src] * ew    (feature width fixed 96)
// block = (32, 8): x = feature lane (covers f, f+32, f+64), y = edge in block
// ---------------------------------------------------------------------------
__global__ void scatter_kernel(const float* __restrict__ r,
                               const int* __restrict__ src,
                               const int* __restrict__ dst,
                               const float* __restrict__ ew,
                               const float* __restrict__ norm,
                               float* __restrict__ acc, int E) {
    int e = blockIdx.x * blockDim.y + threadIdx.y;
    if (e >= E) return;
    int s = src[e];
    int d = dst[e];
    float w = ew[e] * norm[s];
    const float* rs = r + (size_t)s * 96;
    float* ad = acc + (size_t)d * 96;
    int f = threadIdx.x;
    atomicAdd(&ad[f],      rs[f]      * w);
    atomicAdd(&ad[f + 32], rs[f + 32] * w);
    atomicAdd(&ad[f + 64], rs[f + 64] * w);
}

// ---------------------------------------------------------------------------
// WMMA fp32 GEMM:  out[N, ncol] = act( concat(A0|A1|A2)[N, K] @ W[K, ncol] + bias )
// Each A segment is a row-major [N, 96] buffer (lda = 96); segment = k / 96.
// Requires nrows % 80 == 0 (50000 = 625 * 80).
//
// Grid: blockIdx.y = column tile (tn).  The K x 16 weight panel for this tn is
// staged into LDS once per block by the Tensor Data Mover (tensor_load_to_lds),
// then every wave computes a 5-tile (80-row) x 16-col strip with
// V_WMMA_F32_16X16X4_F32, sharing one B fragment across 5 WMMAs.
// ---------------------------------------------------------------------------
__global__ void wmma_gemm_kernel(const float* __restrict__ A0,
                                 const float* __restrict__ A1,
                                 const float* __restrict__ A2,
                                 const float* __restrict__ W,
                                 const float* __restrict__ bias,
                                 float* __restrict__ out,
                                 int nrows, int K, int ncol, int relu) {
    __shared__ float sW[288 * 16];          // up to K=288 rows x 16 cols (18 KB)

    int lane = threadIdx.x & 31;
    int wave = threadIdx.x >> 5;
    int tn   = blockIdx.y;

    // ---- TDM: DMA W[0:K, tn*16 : tn*16+16] into LDS (2-D tile descriptor) ----
    if (wave == 0) {
        unsigned long long gaddr = (unsigned long long)(uintptr_t)(W + (size_t)tn * 16);
        unsigned int ldsoff = (unsigned int)(uintptr_t)&sW[0];   // low 32 bits = LDS offset
        u32x4 g0; i32x8 g1; i32x4 g2; i32x4 g3; i32x8 g4;
        g0[0] = 1u;                                        // count=1, user descriptor
        g0[1] = ldsoff;                                    // lds_addr
        g0[2] = (unsigned int)(gaddr & 0xffffffffull);     // global_addr[31:0]
        g0[3] = (unsigned int)(gaddr >> 32) | (2u << 30);  // global_addr[56:32] | type=2
        g1[0] = (int)(2u << 16);                           // data_size=4B; no multicast/pad
        g1[1] = (int)(((unsigned)ncol & 0xffffu) << 16);   // tensor_dim0[15:0]
        g1[2] = (int)((((unsigned)ncol >> 16) & 0xffffu) | (((unsigned)K & 0xffffu) << 16));
        g1[3] = (int)((((unsigned)K >> 16) & 0xffffu) | (16u << 16)); // td1 hi | tile_dim0=16
        g1[4] = (int)((unsigned)K & 0xffffu);              // tile_dim1=K, tile_dim2=0
        g1[5] = (int)(unsigned)ncol;                       // tensor_dim0_stride = ncol
        g1[6] = 0; g1[7] = 0;                              // dim1_stride unused (2-D)
        g2[0] = 0; g2[1] = 0; g2[2] = 0; g2[3] = 0;
        g3[0] = 0; g3[1] = 0; g3[2] = 0; g3[3] = 0;
        g4[0] = 0; g4[1] = 0; g4[2] = 0; g4[3] = 0;
        g4[4] = 0; g4[5] = 0; g4[6] = 0; g4[7] = 0;
        __builtin_amdgcn_tensor_load_to_lds(g0, g1, g2, g3, g4, 0);
        __builtin_amdgcn_s_wait_tensorcnt(0);
    }
    __syncthreads();

    int ntm    = nrows >> 4;                 // # of 16-row tiles (3125)
    int nstrip = ntm / 5;                    // 5 tiles per wave strip (625)
    // sid is wave-uniform by construction; readfirstlane makes that provable,
    // so the guard is a scalar branch and EXEC stays all-ones for the WMMAs.
    int sid = __builtin_amdgcn_readfirstlane(blockIdx.x * (blockDim.x >> 5) + wave);
    if (sid >= nstrip) return;

    int half = lane >> 4;                    // 0: lanes 0-15, 1: lanes 16-31
    int l16  = lane & 15;
    int koff = half * 2;                     // K sub-offset within the 4-slab
    int col  = tn * 16 + l16;
    int row0 = sid * 80 + l16;               // first A row this lane supplies

    v8f acc[5] = {};

    // One 96-wide K segment against one 96x16 LDS weight panel.
    // A pointer traces directly to a kernel argument on every call path, so
    // loads lower to global_load_b64 (not flat) and cluster ahead of the WMMAs.
    auto run_seg = [&](const float* __restrict__ A, const float* __restrict__ sWk) {
        for (int kin = 0; kin < 96; kin += 4) {
            // B 4x16 fp32 fragment from LDS: lanes 0-15 -> K=0,1 ; 16-31 -> K=2,3
            v2f b;
            b[0] = sWk[(kin + koff) * 16 + l16];
            b[1] = sWk[(kin + koff + 1) * 16 + l16];
            const float* Ab = A + (size_t)row0 * 96 + kin + koff;
#pragma unroll
            for (int t = 0; t < 5; ++t) {
                // A 16x4 fp32 layout: lanes 0-15 -> K=0,1 ; lanes 16-31 -> K=2,3
                v2f a = *(const v2f*)(Ab + (size_t)t * 16 * 96);
                acc[t] = __builtin_amdgcn_wmma_f32_16x16x4_f32(
                    false, a, false, b, (short)0, acc[t], false, false);
            }
        }
    };

    run_seg(A0, sW);
    if (K > 96) {
        run_seg(A1, sW + 96 * 16);
        run_seg(A2, sW + 2 * 96 * 16);
    }

    float bb = bias[col];
#pragma unroll
    for (int t = 0; t < 5; ++t) {
#pragma unroll
        for (int v = 0; v < 8; ++v) {
            int m = sid * 80 + t * 16 + v + half * 8;   // C/D: VGPR v -> M=v (+8 hi half)
            float x = acc[t][v] + bb;
            if (relu) x = fmaxf(x, 0.0f);
            out[(size_t)m * ncol + col] = x;
        }
    }
}

// ---------------------------------------------------------------------------
// Row-wise LayerNorm (+optional ReLU). One wave per row, width w (96 or 32).
// ---------------------------------------------------------------------------
__global__ void ln_kernel(float* __restrict__ x, const float* __restrict__ g,
                          const float* __restrict__ b, int n, int w, int relu) {
    int wave = threadIdx.x >> 5;
    int lane = threadIdx.x & 31;
    int row = blockIdx.x * (blockDim.x >> 5) + wave;
    if (row >= n) return;
    float* xr = x + (size_t)row * w;

    float s = 0.0f;
    for (int f = lane; f < w; f += 32) s += xr[f];
    for (int m = 16; m; m >>= 1) s += __shfl_xor(s, m, 32);
    float mean = s / (float)w;

    float vs = 0.0f;
    for (int f = lane; f < w; f += 32) { float d = xr[f] - mean; vs += d * d; }
    for (int m = 16; m; m >>= 1) vs += __shfl_xor(vs, m, 32);
    float inv = rsqrtf(vs / (float)w + LN_EPS);

    for (int f = lane; f < w; f += 32) {
        float y = (xr[f] - mean) * inv * g[f] + b[f];
        if (relu) y = fmaxf(y, 0.0f);
        xr[f] = y;
    }
}

// ---------------------------------------------------------------------------
// Gather dst_gids rows of h[N, 32] and apply log_softmax. One wave per row.
// ---------------------------------------------------------------------------
__global__ void out_kernel(const float* __restrict__ h, const int* __restrict__ gids,
                           float* __restrict__ out, int M) {
    int wave = threadIdx.x >> 5;
    int lane = threadIdx.x & 31;
    int row = blockIdx.x * (blockDim.x >> 5) + wave;
    if (row >= M) return;
    int gid = gids[row];
    float x = h[(size_t)gid * 32 + lane];
    float mx = x;
    for (int m = 16; m; m >>= 1) mx = fmaxf(mx, __shfl_xor(mx, m, 32));
    float ex = expf(x - mx);
    float s = ex;
    for (int m = 16; m; m >>= 1) s += __shfl_xor(s, m, 32);
    out[(size_t)row * 32 + lane] = x - mx - logf(s);
}

// ---------------------------------------------------------------------------
// Host launcher
// ---------------------------------------------------------------------------
extern "C" void kernel_launch(void* const* d_in, const int* in_sizes, int n_in,
                              void* d_out, int out_size, void* d_ws, size_t ws_size,
                              hipStream_t stream) {
    (void)n_in; (void)out_size; (void)ws_size;

    const float* feat = (const float*)d_in[0];
    const int*   src  = (const int*)d_in[1];
    const int*   dst  = (const int*)d_in[2];
    const float* ew   = (const float*)d_in[3];
    const int*   gids = (const int*)d_in[4];
    const float* W1   = (const float*)d_in[5];
    const float* b1   = (const float*)d_in[6];
    const float* W2   = (const float*)d_in[7];
    const float* b2   = (const float*)d_in[8];
    const float* lng  = (const float*)d_in[9];
    const float* lnb  = (const float*)d_in[10];
    const float* lw[3]  = { (const float*)d_in[11], (const float*)d_in[15], (const float*)d_in[19] };
    const float* lb[3]  = { (const float*)d_in[12], (const float*)d_in[16], (const float*)d_in[20] };
    const float* lg[3]  = { (const float*)d_in[13], (const float*)d_in[17], (const float*)d_in[21] };
    const float* lbe[3] = { (const float*)d_in[14], (const float*)d_in[18], (const float*)d_in[22] };

    const int N = in_sizes[0] / 96;   // 50000 (divisible by 80)
    const int E = in_sizes[1];        // 800000
    const int M = in_sizes[4];        // 10000
    float* out = (float*)d_out;

    // Workspace layout: norm [N] + four [N,96] fp32 buffers (~77 MB total)
    float* wsf = (float*)d_ws;
    size_t nAlign = (size_t)((N + 1023) / 1024) * 1024;
    size_t NF = (size_t)N * 96;
    float* norm = wsf;
    float* B0 = wsf + nAlign;   // h
    float* B1 = B0 + NF;        // r1
    float* B2 = B1 + NF;        // r2
    float* B3 = B2 + NF;        // gemm output / scratch

    const int TPB = 256;
    auto blocks1d = [](long long n, int tpb) { return (int)((n + tpb - 1) / tpb); };

    // ---- degree norm: norm = rsqrt(max(in-degree, 1)) ----
    zero_kernel<<<blocks1d(N, TPB), TPB, 0, stream>>>(norm, N);
    deg_kernel<<<blocks1d(E, TPB), TPB, 0, stream>>>(dst, norm, E);
    norm_finalize_kernel<<<blocks1d(N, TPB), TPB, 0, stream>>>(norm, N);

    auto gemm = [&](const float* A0, const float* A1, const float* A2,
                    const float* W, const float* bias, float* o,
                    int K, int ncol, int relu) {
        int nstrip = (N / 16) / 5;           // N % 80 == 0
        dim3 grid((nstrip + 7) / 8, ncol / 16);
        wmma_gemm_kernel<<<grid, TPB, 0, stream>>>(A0, A1, A2, W, bias, o,
                                                   N, K, ncol, relu);
    };

    // ---- input MLP: h = LN( relu(feat@W1+b1) @ W2 + b2 ) ----
    gemm(feat, feat, feat, W1, b1, B3, 96, 96, /*relu=*/1);
    gemm(B3, B3, B3, W2, b2, B0, 96, 96, /*relu=*/0);
    ln_kernel<<<blocks1d(N, 8), TPB, 0, stream>>>(B0, lng, lnb, N, 96, /*relu=*/0);

    // ---- 3 TAGConv layers (k = 2 hops each) ----
    dim3 sblock(32, 8);
    int sgrid = blocks1d(E, 8);
    for (int l = 0; l < 3; ++l) {
        int ncol = (l == 2) ? 32 : 96;
        // hop 1: r1 = segsum( (h*norm)[src] * ew ) * norm
        zero_kernel<<<blocks1d(NF, TPB), TPB, 0, stream>>>(B1, (int)NF);
        scatter_kernel<<<sgrid, sblock, 0, stream>>>(B0, src, dst, ew, norm, B1, E);
        row_scale_kernel<<<blocks1d(NF, TPB), TPB, 0, stream>>>(B1, norm, (int)NF);
        // hop 2: r2 = segsum( (r1*norm)[src] * ew ) * norm
        zero_kernel<<<blocks1d(NF, TPB), TPB, 0, stream>>>(B2, (int)NF);
        scatter_kernel<<<sgrid, sblock, 0, stream>>>(B1, src, dst, ew, norm, B2, E);
        row_scale_kernel<<<blocks1d(NF, TPB), TPB, 0, stream>>>(B2, norm, (int)NF);
        // fused concat-GEMM: B3 = [h|r1|r2] @ lw + lb, then LN (+ReLU except last)
        gemm(B0, B1, B2, lw[l], lb[l], B3, 288, ncol, /*relu=*/0);
        ln_kernel<<<blocks1d(N, 8), TPB, 0, stream>>>(B3, lg[l], lbe[l], N, ncol, l < 2);
        std::swap(B0, B3);
    }

    // ---- gather + log_softmax ----
    out_kernel<<<blocks1d(M, 8), TPB, 0, stream>>>(B0, gids, out, M);
}